// ParallelTransportUnpool_87582973100651
// MI455X (gfx1250) — compile-verified
//
#include <hip/hip_runtime.h>
#include <stdint.h>

// ---------------------------------------------------------------------------
// ParallelTransportUnpool for MI455X (gfx1250).
// Pure bandwidth problem: ~819 MB store + ~102 MB effective read (x fits in
// 192 MB L2). One wave32 per edge; b128 loads, non-temporal b128 stores.
// gfx1250 async global<->LDS DMA path for the pure-copy half (ASYNCcnt),
// with TH_NT on the output stream so x stays L2-resident.
// ---------------------------------------------------------------------------

typedef __attribute__((ext_vector_type(4))) float v4f;
typedef int v4i __attribute__((vector_size(16)));   // element type the async
                                                    // builtins expect

#define C_DIM 128
#define ROW   (2 * C_DIM * 2)   // 512 floats per node row
#define ROWV  (ROW / 4)         // 128 float4 chunks per row
#define WPB   8                 // waves (edges) per 256-thread block
#define BLOCK (32 * WPB)

// gfx12+ CPol: TH in bits [2:0]; TH_NT == 1 (non-temporal).
#define CPOL_RT 0
#define CPOL_NT 1

#ifndef __has_builtin
#define __has_builtin(x) 0
#endif

#if __has_builtin(__builtin_amdgcn_global_load_async_to_lds_b128) && \
    __has_builtin(__builtin_amdgcn_global_store_async_from_lds_b128) && \
    __has_builtin(__builtin_amdgcn_s_wait_asynccnt)
#define HAVE_ASYNC_LDS 1
#else
#define HAVE_ASYNC_LDS 0
#endif

// Casts: generic pointers -> addrspace-qualified int4 pointers for the
// async builtins (they take non-const v4i* in AS1/AS3).
#define GAS1(p) ((__attribute__((address_space(1))) v4i*)(p))
#define LAS3(p) ((__attribute__((address_space(3))) v4i*)(p))

__global__ void build_map_kernel(const int* __restrict__ nodes,
                                 int* __restrict__ map, int n_coarse) {
  int i = blockIdx.x * blockDim.x + threadIdx.x;
  if (i < n_coarse) map[nodes[i]] = i;   // inverse permutation (scatter)
}

template <bool USE_MAP>
__global__ __launch_bounds__(BLOCK) void unpool_kernel(
    const float* __restrict__ x,
    const float* __restrict__ conn,
    const int* __restrict__ edge_src,
    const int* __restrict__ edge_dst,
    const int* __restrict__ map_or_nodes,   // map (USE_MAP) or sorted nodes
    int n_coarse,
    float* __restrict__ out,
    int n_edges) {
  const int lane = threadIdx.x & 31;
  const int wave = threadIdx.x >> 5;
  const int e = blockIdx.x * WPB + wave;
  if (e >= n_edges) return;               // wave-uniform guard (EXEC stays full)

  // --- wave-uniform edge metadata; force into SGPRs ------------------------
  const int src = __builtin_amdgcn_readfirstlane(edge_src[e]);
  const int dst = __builtin_amdgcn_readfirstlane(edge_dst[e]);

  int ci;
  if (USE_MAP) {
    ci = __builtin_amdgcn_readfirstlane(map_or_nodes[src]);
  } else {
    // sorted-array inverse lookup (nodes[ci] == src guaranteed by construction)
    const int* __restrict__ nodes = map_or_nodes;
    int lo = 0, hi = n_coarse - 1;
    while (lo < hi) {
      int mid = (lo + hi + 1) >> 1;
      if (nodes[mid] <= src) lo = mid; else hi = mid - 1;
    }
    ci = __builtin_amdgcn_readfirstlane(lo);
  }

  const float c = __uint_as_float(
      __builtin_amdgcn_readfirstlane(__float_as_uint(conn[2 * e])));
  const float s = __uint_as_float(
      __builtin_amdgcn_readfirstlane(__float_as_uint(conn[2 * e + 1])));

  const v4f* __restrict__ xrow = (const v4f*)(x + (size_t)ci * ROW);
  v4f* __restrict__ orow = (v4f*)(out + (size_t)dst * ROW);

  // --- half 0 (chunks 0..63): straight 1KB copy ----------------------------
#if HAVE_ASYNC_LDS
  // gfx1250 async DMA: global -> LDS, then LDS -> global; no VGPR round-trip.
  __shared__ v4f stage[WPB][ROWV / 2];
  v4f* lbuf = &stage[wave][0];

  __builtin_amdgcn_global_load_async_to_lds_b128(GAS1(xrow + lane),
                                                 LAS3(lbuf + lane), 0, CPOL_RT);
  __builtin_amdgcn_global_load_async_to_lds_b128(GAS1(xrow + lane + 32),
                                                 LAS3(lbuf + lane + 32), 0,
                                                 CPOL_RT);
#else
#pragma unroll
  for (int j = 0; j < 2; ++j) {
    v4f v = xrow[j * 32 + lane];
    __builtin_nontemporal_store(v, &orow[j * 32 + lane]);
  }
#endif

  // --- half 1 (chunks 64..127): complex rotate by conj(e^{i theta}) --------
  // layout per chunk: {re0, im0, re1, im1}
#pragma unroll
  for (int j = 2; j < 4; ++j) {
    v4f v = xrow[j * 32 + lane];
    v4f r;
    r.x = v.x * c + v.y * s;   // re' = re*cos + im*sin
    r.y = v.y * c - v.x * s;   // im' = im*cos - re*sin
    r.z = v.z * c + v.w * s;
    r.w = v.w * c - v.z * s;
    __builtin_nontemporal_store(r, &orow[j * 32 + lane]);
  }

#if HAVE_ASYNC_LDS
  // ensure async loads have landed in LDS before async stores read it
  __builtin_amdgcn_s_wait_asynccnt(0);
  // NT on the output stream: 819 MB written once, never reread -> keep it
  // from evicting x (102 MB, reread ~4x) out of the 192 MB L2.
  __builtin_amdgcn_global_store_async_from_lds_b128(GAS1(orow + lane),
                                                    LAS3(lbuf + lane), 0,
                                                    CPOL_NT);
  __builtin_amdgcn_global_store_async_from_lds_b128(GAS1(orow + lane + 32),
                                                    LAS3(lbuf + lane + 32), 0,
                                                    CPOL_NT);
  // S_ENDPGM performs an implicit wait-idle, covering outstanding async stores.
#endif
}

extern "C" void kernel_launch(void* const* d_in, const int* in_sizes, int n_in,
                              void* d_out, int out_size, void* d_ws,
                              size_t ws_size, hipStream_t stream) {
  const float* x     = (const float*)d_in[0];  // (n_coarse, 2, 128, 2) f32
  const float* conn  = (const float*)d_in[1];  // (n_edges, 2) f32
  const int*   nodes = (const int*)d_in[2];    // (n_coarse,) sorted
  const int*   edges = (const int*)d_in[3];    // (2, n_edges)
  // d_in[4] = num_nodes scalar on device; derive from out_size instead.

  const int n_coarse = in_sizes[2];
  const int n_edges  = in_sizes[3] / 2;
  const int* edge_src = edges;
  const int* edge_dst = edges + n_edges;
  const int num_nodes = out_size / ROW;

  float* out = (float*)d_out;
  const int grid = (n_edges + WPB - 1) / WPB;

  if (ws_size >= (size_t)num_nodes * sizeof(int)) {
    int* map = (int*)d_ws;
    build_map_kernel<<<(n_coarse + 255) / 256, 256, 0, stream>>>(nodes, map,
                                                                 n_coarse);
    unpool_kernel<true><<<grid, BLOCK, 0, stream>>>(
        x, conn, edge_src, edge_dst, map, n_coarse, out, n_edges);
  } else {
    unpool_kernel<false><<<grid, BLOCK, 0, stream>>>(
        x, conn, edge_src, edge_dst, nodes, n_coarse, out, n_edges);
  }
}